// KnotEntangle_83348135346233
// MI455X (gfx1250) — compile-verified
//
#include <hip/hip_runtime.h>

// ---------------------------------------------------------------------------
// KnotEntangle restructured for MI455X (gfx1250, wave32, WMMA f32 16x16x4).
//
// Math restructure (exact, uses linearity of irfft and the (i,j) double sum):
//   result[b,s] = irfft(Acc[b,:])[s] + sum_i wsmear[b,i]*smear[b,i,s]
//   Acc[b,f]    = sum_j cos(pol_j)*sig[b,j,f]*u0[b,j,f]
//                     + sin(pol_j)*w0[b,j,f]*u1[b,j,f]          (complex)
//   u0[b,j,:]   = polk[j]   @ w0[b,j,:]     -> WMMA GEMM-0 (stacked [Pr;Pi])
//   u1[b,j,:]   = polk[j]^T @ sig[b,j,:]    -> WMMA GEMM-1 (sig rows x [Pr|Pi])
//   w0[b,j,g]   = sum_{i!=j} mix[b,i,j]*sig[b,i,g]
//   wsmear[b,i] = sum_{j!=i} (1 - mix[b,i,j])
//   mix uses only sig[...,0] = sum_s smear (DC bin, real).
// Forward rDFT is a WMMA GEMM against integer-exact twiddle tables, with the
// table panel double-buffered through LDS via global_load_async_to_lds_b128.
// ---------------------------------------------------------------------------

typedef float v2f __attribute__((ext_vector_type(2)));
typedef float v8f __attribute__((ext_vector_type(8)));

constexpr int   B  = 16, K = 32, S = 1024, C = 8, F = 513;
constexpr int   FP = 528;           // F padded to multiple of 16
constexpr int   BK = B * K;         // 512 rows for the forward-DFT GEMM
constexpr int   M2F  = 2 * F;       // stacked [Pr;Pi] rows = 1026
constexpr int   MPAD = 1040;        // 65 * 16
constexpr int   CK   = 32;          // k_dft LDS staging chunk (rows of table)
constexpr float PI2  = 6.28318530717958647692f;

// workspace offsets, in floats
constexpr size_t WS_COS   = 0;                                  // [S][FP]
constexpr size_t WS_SIN   = WS_COS   + (size_t)S * FP;          // [S][FP]
constexpr size_t WS_SMEAR = WS_SIN   + (size_t)S * FP;          // [BK][S]
constexpr size_t WS_SIGRE = WS_SMEAR + (size_t)BK * S;          // [BK][FP]
constexpr size_t WS_SIGIM = WS_SIGRE + (size_t)BK * FP;         // [BK][FP]
constexpr size_t WS_MIX   = WS_SIGIM + (size_t)BK * FP;         // [B][K][K] (i,j)
constexpr size_t WS_WSM   = WS_MIX   + (size_t)B * K * K;       // [B][K]
constexpr size_t WS_W0RE  = WS_WSM   + (size_t)B * K;           // [B][K][FP]
constexpr size_t WS_W0IM  = WS_W0RE  + (size_t)BK * FP;         // [B][K][FP]
constexpr size_t WS_G0    = WS_W0IM  + (size_t)BK * FP;         // [K][MPAD][32]
constexpr size_t WS_G1    = WS_G0    + (size_t)K * MPAD * 32;   // [K][2][32][FP]
constexpr size_t WS_U0RE  = WS_G1    + (size_t)K * 2 * 32 * FP; // [B][K][FP]
constexpr size_t WS_U0IM  = WS_U0RE  + (size_t)BK * FP;
constexpr size_t WS_U1RE  = WS_U0IM  + (size_t)BK * FP;
constexpr size_t WS_U1IM  = WS_U1RE  + (size_t)BK * FP;
constexpr size_t WS_ACCRE = WS_U1IM  + (size_t)BK * FP;         // [B][FP]
constexpr size_t WS_ACCIM = WS_ACCRE + (size_t)B * FP;          // [B][FP]
constexpr size_t WS_TOTAL = (WS_ACCIM + (size_t)B * FP) * sizeof(float);

// ---------------------------------------------------------------------------
// DFT twiddle tables: cosT[s][f] = cos(2*pi*s*f/S), sinT likewise.
// Integer (s*f) mod S keeps the phase argument exact. f >= F columns -> 0,
// which auto-zero-pads the GEMM outputs for padded bins.
__global__ void k_tables(float* ws) {
  int idx = blockIdx.x * 256 + threadIdx.x;
  if (idx >= S * FP) return;
  int s = idx / FP, f = idx % FP;
  float cs = 0.f, sn = 0.f;
  if (f < F) {
    int m = (s * f) & (S - 1);
    float ang = PI2 * (float)m / (float)S;
    sincosf(ang, &sn, &cs);
  }
  ws[WS_COS + idx] = cs;
  ws[WS_SIN + idx] = sn;
}

// smear[b,k,s] = sum_c amp*cos(freq*t + phase), t affine in (s+1)/S
__global__ void k_smear(const float* __restrict__ x, const float* __restrict__ fr,
                        const float* __restrict__ ph, const float* __restrict__ am,
                        const float* __restrict__ win, float* ws) {
  int bk = blockIdx.x;
  int k  = bk % K;
  float lower = win[0], upper = win[1];
  float xv    = x[bk];
  float xstep = (upper - lower) * xv / (float)S;
  float xlow  = (1.f - lower) * xv;
  float fv[C], pv[C], av[C];
  #pragma unroll
  for (int c = 0; c < C; ++c) { fv[c] = fr[k*C+c]; pv[c] = ph[k*C+c]; av[c] = am[k*C+c]; }
  float* sm = ws + WS_SMEAR + (size_t)bk * S;
  for (int s = threadIdx.x; s < S; s += 256) {
    float t = xstep * ((float)(s + 1) / (float)S) + xlow;
    float acc = 0.f;
    #pragma unroll
    for (int c = 0; c < C; ++c) acc += av[c] * cosf(fv[c] * t + pv[c]);
    sm[s] = acc;
  }
}

// Forward rDFT as f32 WMMA GEMM: sig[bk][f] = sum_s smear[bk][s]*table[s][f].
// One wave per 16x16 tile; K = 1024 in steps of 4 (v_wmma_f32_16x16x4_f32).
// The 16-column table panel is double-buffered through LDS with
// global_load_async_to_lds_b128 (4 issues x 32 lanes x 16B = one 32-row chunk),
// ordered by s_wait_asynccnt (single wave per block, so no barrier needed).
__global__ void __launch_bounds__(32) k_dft(float* ws) {
  __shared__ float buf[2][CK * 16];
  int tm = blockIdx.x, tn = blockIdx.y, plane = blockIdx.z;
  int l  = threadIdx.x;
  const float* A = ws + WS_SMEAR;
  const float* T = ws + (plane ? WS_SIN : WS_COS);
  int mrow  = tm * 16 + (l & 15);            // A fragment row
  int lcol  = (l & 15);                      // B fragment col within tile
  int khalf = (l >> 4) * 2;                  // K split across half-waves
  const float* arow = A + (size_t)mrow * S;

  int r = l >> 2, q = l & 3;                 // staging: 8 rows x 4 quarters/issue
  auto stage = [&](int c, int k0) {
    unsigned lbase = (unsigned)(uintptr_t)(&buf[c][0]);
    #pragma unroll
    for (int i = 0; i < 4; ++i) {
      int row = i * 8 + r;
      const float* g = T + (size_t)(k0 + row) * FP + tn * 16 + q * 4;
      unsigned loff = lbase + (unsigned)(row * 16 + q * 4) * 4u;
      asm volatile("global_load_async_to_lds_b128 %0, %1, off"
                   :: "v"(loff), "v"(g) : "memory");
    }
  };

  stage(0, 0);
  v8f c = {};
  for (int k0 = 0; k0 < S; k0 += CK) {
    int cur = (k0 / CK) & 1;
    if (k0 + CK < S) {
      stage(cur ^ 1, k0 + CK);
      asm volatile("s_wait_asynccnt 0x4" ::: "memory");  // chunk `cur` landed
    } else {
      asm volatile("s_wait_asynccnt 0x0" ::: "memory");
    }
    #pragma unroll
    for (int kk = 0; kk < CK; kk += 4) {
      v2f a = *(const v2f*)(arow + k0 + kk + khalf);      // 8B aligned
      v2f b;
      b.x = buf[cur][(kk + khalf)     * 16 + lcol];
      b.y = buf[cur][(kk + khalf + 1) * 16 + lcol];
      c = __builtin_amdgcn_wmma_f32_16x16x4_f32(false, a, false, b,
                                                (short)0, c, false, false);
    }
  }
  float sgn  = plane ? -1.f : 1.f;           // rfft: im = -sum x*sin
  float* out = ws + (plane ? WS_SIGIM : WS_SIGRE);
  int ncol   = tn * 16 + lcol;
  int mbase  = tm * 16 + (l >> 4) * 8;       // C/D layout: M = v + (l>>4)*8
  #pragma unroll
  for (int v = 0; v < 8; ++v)
    out[(size_t)(mbase + v) * FP + ncol] = sgn * c[v];
}

// mix[b,i,j] = irregular_gauss(dc_i*dc_j/S, params[j]); dc = sig[...,0]
__global__ void k_mix(const float* __restrict__ gm, const float* __restrict__ gl,
                      const float* __restrict__ gh, float* ws) {
  int idx = blockIdx.x * 256 + threadIdx.x;
  if (idx >= B * K * K) return;
  int j = idx % K, i = (idx / K) % K, b = idx / (K * K);
  float dci = ws[WS_SIGRE + (size_t)(b*K + i) * FP];
  float dcj = ws[WS_SIGRE + (size_t)(b*K + j) * FP];
  float corr = dci * dcj / (float)S;
  float sd = (corr < gm[j]) ? expf(gl[j]) : expf(gh[j]);
  float d  = (corr - gm[j]) / sd;
  ws[WS_MIX + idx] = expf(-0.5f * d * d);
}

__global__ void k_wsmear(float* ws) {
  int idx = blockIdx.x * 256 + threadIdx.x;
  if (idx >= B * K) return;
  int i = idx % K, b = idx / K;
  float a = 0.f;
  for (int j = 0; j < K; ++j)
    if (j != i) a += 1.f - ws[WS_MIX + (size_t)b*K*K + i*K + j];
  ws[WS_WSM + idx] = a;
}

// w0[b,j,f] = sum_{i!=j} mix[b,i,j]*sig[b,i,f]   (complex; pads stay zero)
__global__ void k_wsum0(float* ws) {
  int idx = blockIdx.x * 256 + threadIdx.x;
  if (idx >= BK * FP) return;
  int f = idx % FP, j = (idx / FP) % K, b = idx / (FP * K);
  float ar = 0.f, ai = 0.f;
  for (int i = 0; i < K; ++i) {
    if (i == j) continue;
    float m = ws[WS_MIX + (size_t)b*K*K + i*K + j];
    ar += m * ws[WS_SIGRE + (size_t)(b*K + i) * FP + f];
    ai += m * ws[WS_SIGIM + (size_t)(b*K + i) * FP + f];
  }
  ws[WS_W0RE + idx] = ar;
  ws[WS_W0IM + idx] = ai;
}

// GEMM-0 per j: [Pr;Pi] (2F x F) @ W (F x 32), W cols = 16 b x {w0re,w0im}.
// K = 513: 128 x K4 WMMA steps + rank-1 VALU fixup for g = 512.
// M tail (rows 1026..1039) row-clamped on load; garbage rows never read back.
__global__ void __launch_bounds__(32) k_gemm0(const float* __restrict__ pre,
                                              const float* __restrict__ pim,
                                              float* ws) {
  int tm = blockIdx.x, tn = blockIdx.y, j = blockIdx.z;
  int l  = threadIdx.x;
  int mg = tm * 16 + (l & 15);
  int mc = (mg < M2F) ? mg : (M2F - 1);
  const float* arow = (mc < F) ? (pre + (size_t)j*F*F + (size_t)mc * F)
                               : (pim + (size_t)j*F*F + (size_t)(mc - F) * F);
  int ng = tn * 16 + (l & 15);               // 0..31: plane = ng>>4, batch = ng&15
  const float* bcol = ws + (ng < 16 ? WS_W0RE : WS_W0IM)
                         + (size_t)(ng & 15) * K * FP + (size_t)j * FP;
  int khalf = (l >> 4) * 2;
  v8f c = {};
  for (int g0 = 0; g0 < 512; g0 += 4) {
    v2f a; a.x = arow[g0 + khalf]; a.y = arow[g0 + khalf + 1];  // F odd: scalar
    v2f b; b.x = bcol[g0 + khalf]; b.y = bcol[g0 + khalf + 1];
    c = __builtin_amdgcn_wmma_f32_16x16x4_f32(false, a, false, b,
                                              (short)0, c, false, false);
  }
  float bval   = bcol[512];                  // rank-1 tail: B[512][ng]
  int   mbase  = tm * 16 + (l >> 4) * 8;
  float* g0out = ws + WS_G0 + (size_t)j * MPAD * 32;
  #pragma unroll
  for (int v = 0; v < 8; ++v) {
    int m   = mbase + v;
    int mcv = (m < M2F) ? m : (M2F - 1);
    float aval = (mcv < F) ? pre[(size_t)j*F*F + (size_t)mcv * F + 512]
                           : pim[(size_t)j*F*F + (size_t)(mcv - F) * F + 512];
    g0out[(size_t)m * 32 + ng] = c[v] + aval * bval;
  }
}

// GEMM-1 per j: W (32 x F) @ P-plane (F x F), W rows = 16 b x {sigre,sigim},
// plane in {Pr,Pi}. Reads polk twice, once per row tile (vs 16x for the VALU matvec).
// N tail columns clamped to F-1 on load; pad columns stored but never read.
__global__ void __launch_bounds__(32) k_gemm1(const float* __restrict__ pre,
                                              const float* __restrict__ pim,
                                              float* ws) {
  int tm = blockIdx.x;                       // 0..1 : rows 0-15 = sr, 16-31 = si
  int ty = blockIdx.y;                       // 0..65 : plane = ty/33, tnf = ty%33
  int j  = blockIdx.z;
  int plane = ty / 33, tnf = ty % 33;
  int l  = threadIdx.x;
  int mg = tm * 16 + (l & 15);
  const float* arow = ws + (mg < 16 ? WS_SIGRE : WS_SIGIM)
                         + (size_t)((mg & 15) * K + j) * FP;
  int ncol = tnf * 16 + (l & 15);
  int nc   = (ncol < F) ? ncol : (F - 1);
  const float* Pp = (plane ? pim : pre) + (size_t)j * F * F;
  int khalf = (l >> 4) * 2;
  v8f c = {};
  for (int g0 = 0; g0 < 512; g0 += 4) {
    v2f a = *(const v2f*)(arow + g0 + khalf);                   // 8B aligned
    v2f b;
    b.x = Pp[(size_t)(g0 + khalf)     * F + nc];
    b.y = Pp[(size_t)(g0 + khalf + 1) * F + nc];
    c = __builtin_amdgcn_wmma_f32_16x16x4_f32(false, a, false, b,
                                              (short)0, c, false, false);
  }
  float bval = Pp[(size_t)512 * F + nc];     // rank-1 tail: g = 512
  int mbase  = tm * 16 + (l >> 4) * 8;
  float* g1  = ws + WS_G1 + ((size_t)j * 2 + plane) * 32 * FP;
  #pragma unroll
  for (int v = 0; v < 8; ++v) {
    int m = mbase + v;
    float aval = ws[(m < 16 ? WS_SIGRE : WS_SIGIM)
                    + (size_t)((m & 15) * K + j) * FP + 512];
    g1[(size_t)m * FP + ncol] = c[v] + aval * bval;
  }
}

// Assemble complex u0 = (Pr+iPi)(w0r+iw0i) and u1 = (sr+isi)^T(Pr+iPi)
// from the real GEMM planes.
__global__ void k_comb(float* ws) {
  int idx = blockIdx.x * 256 + threadIdx.x;
  if (idx >= BK * FP) return;
  int f = idx % FP, j = (idx / FP) % K, b = idx / (FP * K);
  float u0r = 0.f, u0i = 0.f, u1r = 0.f, u1i = 0.f;
  if (f < F) {
    const float* g0 = ws + WS_G0 + (size_t)j * MPAD * 32;
    float prr = g0[(size_t)f       * 32 + b];        // Pr @ w0r
    float pri = g0[(size_t)f       * 32 + 16 + b];   // Pr @ w0i
    float pir = g0[(size_t)(F + f) * 32 + b];        // Pi @ w0r
    float pii = g0[(size_t)(F + f) * 32 + 16 + b];   // Pi @ w0i
    u0r = prr - pii;
    u0i = pri + pir;
    const float* g1 = ws + WS_G1 + (size_t)j * 2 * 32 * FP;
    float srPr = g1[(size_t)b        * FP + f];      // plane 0, row b
    float siPr = g1[(size_t)(16 + b) * FP + f];      // plane 0, row 16+b
    float srPi = g1[(size_t)(32 + b) * FP + f];      // plane 1, row b
    float siPi = g1[(size_t)(48 + b) * FP + f];      // plane 1, row 16+b
    u1r = srPr - siPi;
    u1i = srPi + siPr;
  }
  ws[WS_U0RE + idx] = u0r;
  ws[WS_U0IM + idx] = u0i;
  ws[WS_U1RE + idx] = u1r;
  ws[WS_U1IM + idx] = u1i;
}

// Acc[b,f] = sum_j cos(pol_j)*sig*u0 + sin(pol_j)*w0*u1  (complex products)
__global__ void k_acc(const float* __restrict__ pol, float* ws) {
  int idx = blockIdx.x * 256 + threadIdx.x;
  if (idx >= B * FP) return;
  int f = idx % FP, b = idx / FP;
  float ar = 0.f, ai = 0.f;
  if (f < F) {
    for (int j = 0; j < K; ++j) {
      float cp, sp; sincosf(pol[j], &sp, &cp);
      size_t o = (size_t)(b*K + j) * FP + f;
      float sr  = ws[WS_SIGRE + o], si  = ws[WS_SIGIM + o];
      float u0r = ws[WS_U0RE + o],  u0i = ws[WS_U0IM + o];
      float w0r = ws[WS_W0RE + o],  w0i = ws[WS_W0IM + o];
      float u1r = ws[WS_U1RE + o],  u1i = ws[WS_U1IM + o];
      ar += cp * (sr * u0r - si * u0i) + sp * (w0r * u1r - w0i * u1i);
      ai += cp * (sr * u0i + si * u0r) + sp * (w0r * u1i + w0i * u1r);
    }
  }
  ws[WS_ACCRE + (size_t)b * FP + f] = ar;
  ws[WS_ACCIM + (size_t)b * FP + f] = ai;
}

// irfft of Acc (16 rows) + smear passthrough term + gaussian attention gating.
__global__ void k_final(const float* __restrict__ x,  const float* __restrict__ win,
                        const float* __restrict__ aw, const float* __restrict__ ab,
                        const float* __restrict__ asc, float* ws,
                        float* __restrict__ out) {
  int idx = blockIdx.x * 256 + threadIdx.x;
  if (idx >= B * S) return;
  int s = idx % S, b = idx / S;
  const float* ct = ws + WS_COS   + (size_t)s * FP;
  const float* st = ws + WS_SIN   + (size_t)s * FP;
  const float* ar = ws + WS_ACCRE + (size_t)b * FP;
  const float* ai = ws + WS_ACCIM + (size_t)b * FP;
  // x[s] = (1/S) * sum_f w_f * Re(Acc[f] e^{+i2pi f s/S}); w_0 = w_{S/2} = 1
  float ent = ar[0] * ct[0] - ai[0] * st[0];
  ent += ar[F-1] * ct[F-1] - ai[F-1] * st[F-1];
  for (int f = 1; f < F - 1; ++f)
    ent += 2.f * (ar[f] * ct[f] - ai[f] * st[f]);
  ent *= (1.f / (float)S);
  float smt = 0.f;
  for (int i = 0; i < K; ++i)
    smt += ws[WS_WSM + b*K + i] * ws[WS_SMEAR + (size_t)(b*K + i) * S + s];
  float lower = win[0], upper = win[1];
  float mm = 0.f;
  for (int k = 0; k < K; ++k) mm += x[b*K + k];
  mm *= (1.f / (float)K);
  float xit = (float)(s + 1) / (float)S;
  float gsum = 0.f;
  for (int k = 0; k < K; ++k) {
    float am = x[b*K + k] * aw[k] + ab[k];
    float lo = (1.f - lower * asc[k]) * mm;
    float hi = (1.f + upper * asc[k]) * mm;
    float g  = (hi - lo) * xit + lo;
    float sd = (g < am) ? expf(lo) : expf(hi);
    float d  = (g - am) / sd;
    gsum += expf(-0.5f * d * d);
  }
  out[idx] = gsum * (ent + smt);
}

extern "C" void kernel_launch(void* const* d_in, const int* in_sizes, int n_in,
                              void* d_out, int out_size, void* d_ws, size_t ws_size,
                              hipStream_t stream) {
  const float* x   = (const float*)d_in[0];
  const float* kfr = (const float*)d_in[1];
  const float* kph = (const float*)d_in[2];
  const float* kam = (const float*)d_in[3];
  const float* win = (const float*)d_in[4];
  const float* gm  = (const float*)d_in[5];
  const float* gl  = (const float*)d_in[6];
  const float* gh  = (const float*)d_in[7];
  const float* pol = (const float*)d_in[8];
  const float* pre = (const float*)d_in[9];
  const float* pim = (const float*)d_in[10];
  const float* aw  = (const float*)d_in[11];
  const float* ab  = (const float*)d_in[12];
  const float* asc = (const float*)d_in[13];
  float* out = (float*)d_out;
  float* ws  = (float*)d_ws;
  if (ws_size < WS_TOTAL) return;

  k_tables <<<(S * FP + 255) / 256, 256, 0, stream>>>(ws);
  k_smear  <<<BK, 256, 0, stream>>>(x, kfr, kph, kam, win, ws);
  k_dft    <<<dim3(BK / 16, FP / 16, 2), 32, 0, stream>>>(ws);
  k_mix    <<<(B * K * K + 255) / 256, 256, 0, stream>>>(gm, gl, gh, ws);
  k_wsmear <<<(B * K + 255) / 256, 256, 0, stream>>>(ws);
  k_wsum0  <<<(BK * FP + 255) / 256, 256, 0, stream>>>(ws);
  k_gemm0  <<<dim3(MPAD / 16, 2, K), 32, 0, stream>>>(pre, pim, ws);
  k_gemm1  <<<dim3(2, 66, K), 32, 0, stream>>>(pre, pim, ws);
  k_comb   <<<(BK * FP + 255) / 256, 256, 0, stream>>>(ws);
  k_acc    <<<(B * FP + 255) / 256, 256, 0, stream>>>(pol, ws);
  k_final  <<<(B * S + 255) / 256, 256, 0, stream>>>(x, win, aw, ab, asc, ws, out);
}